// GAT_79078937854206
// MI455X (gfx1250) — compile-verified
//
#include <hip/hip_runtime.h>
#include <stdint.h>

typedef __attribute__((ext_vector_type(2))) float v2f;
typedef __attribute__((ext_vector_type(4))) float v4f;
typedef __attribute__((ext_vector_type(8))) float v8f;

#define NEG_SLOPE 0.2f
#define NCLS 40

// ---- helpers -------------------------------------------------------------
__device__ __forceinline__ unsigned fkey(float f) {
  unsigned b = __float_as_uint(f);
  return (b & 0x80000000u) ? ~b : (b | 0x80000000u);
}
__device__ __forceinline__ float funkey(unsigned k) {
  return (k & 0x80000000u) ? __uint_as_float(k & 0x7FFFFFFFu)
                           : __uint_as_float(~k);
}
__device__ __forceinline__ float lrelu(float x) {
  return x > 0.f ? x : NEG_SLOPE * x;
}
__device__ __forceinline__ void edge_nodes(const long long* __restrict__ ei,
                                           long long E, long long e,
                                           int* s, int* d) {
  if (e < E) { *s = (int)ei[e]; *d = (int)ei[E + e]; }
  else       { *s = *d = (int)(e - E); }            // self loop
}

// ---- zero ----------------------------------------------------------------
__global__ void k_zero(float* __restrict__ p, long long n) {
  long long i = (long long)blockIdx.x * blockDim.x + threadIdx.x;
  long long stride = (long long)gridDim.x * blockDim.x;
  for (; i < n; i += stride) p[i] = 0.f;
}

// ---- GEMM1: h1[N,64] = x[N,512] @ W1[512,64]  (fp32 WMMA 16x16x4) -------
#define W1T_STRIDE 518   // even (8B-aligned b64 LDS loads), 518%64=6 -> conflict-free
__global__ void __launch_bounds__(256) k_gemm1(const float* __restrict__ x,
                                               const float* __restrict__ W1,
                                               float* __restrict__ h1, int N) {
  __shared__ float w1t[64 * W1T_STRIDE];
  int tid = threadIdx.x;
  for (int i = tid; i < 512 * 64; i += 256) {   // stage W1 transposed
    int k = i >> 6, n = i & 63;
    w1t[n * W1T_STRIDE + k] = W1[i];
  }
  __syncthreads();

  int wave = tid >> 5, lane = tid & 31;
  int tile = blockIdx.x * 8 + wave;
  if (tile * 16 >= N) return;
  int row0 = tile * 16;
  int m    = lane & 15;
  int koff = (lane >> 4) << 1;                  // K sub-offset per half-wave

  const float* ap = x + (long long)(row0 + m) * 512 + koff;
  const float* b0p = &w1t[( 0 + m) * W1T_STRIDE + koff];
  const float* b1p = &w1t[(16 + m) * W1T_STRIDE + koff];
  const float* b2p = &w1t[(32 + m) * W1T_STRIDE + koff];
  const float* b3p = &w1t[(48 + m) * W1T_STRIDE + koff];

  v8f acc0 = {0,0,0,0,0,0,0,0}, acc1 = {0,0,0,0,0,0,0,0};
  v8f acc2 = {0,0,0,0,0,0,0,0}, acc3 = {0,0,0,0,0,0,0,0};

#pragma unroll 4
  for (int kk = 0; kk < 512; kk += 4) {
    v2f a  = *reinterpret_cast<const v2f*>(ap  + kk);
    v2f b0 = *reinterpret_cast<const v2f*>(b0p + kk);
    v2f b1 = *reinterpret_cast<const v2f*>(b1p + kk);
    v2f b2 = *reinterpret_cast<const v2f*>(b2p + kk);
    v2f b3 = *reinterpret_cast<const v2f*>(b3p + kk);
    acc0 = __builtin_amdgcn_wmma_f32_16x16x4_f32(false, a, false, b0, (short)0, acc0, false, false);
    acc1 = __builtin_amdgcn_wmma_f32_16x16x4_f32(false, a, false, b1, (short)0, acc1, false, false);
    acc2 = __builtin_amdgcn_wmma_f32_16x16x4_f32(false, a, false, b2, (short)0, acc2, false, false);
    acc3 = __builtin_amdgcn_wmma_f32_16x16x4_f32(false, a, false, b3, (short)0, acc3, false, false);
  }

  int rbase = row0 + ((lane >> 4) << 3);
#pragma unroll
  for (int v = 0; v < 8; v++) {
    float* o = h1 + (long long)(rbase + v) * 64 + m;
    o[0] = acc0[v]; o[16] = acc1[v]; o[32] = acc2[v]; o[48] = acc3[v];
  }
}

// ---- GEMM2: h2[N,40] = hin[N,64] @ W2[64,40] ----------------------------
#define W2T_STRIDE 66
__global__ void __launch_bounds__(256) k_gemm2(const float* __restrict__ hin,
                                               const float* __restrict__ W2,
                                               float* __restrict__ h2, int N) {
  __shared__ float w2t[48 * W2T_STRIDE];
  int tid = threadIdx.x;
  for (int i = tid; i < 48 * 64; i += 256) {
    int n = i >> 6, k = i & 63;
    w2t[n * W2T_STRIDE + k] = (n < NCLS) ? W2[k * NCLS + n] : 0.f;
  }
  __syncthreads();

  int wave = tid >> 5, lane = tid & 31;
  int tile = blockIdx.x * 8 + wave;
  if (tile * 16 >= N) return;
  int row0 = tile * 16;
  int m    = lane & 15;
  int koff = (lane >> 4) << 1;

  const float* ap  = hin + (long long)(row0 + m) * 64 + koff;
  const float* b0p = &w2t[( 0 + m) * W2T_STRIDE + koff];
  const float* b1p = &w2t[(16 + m) * W2T_STRIDE + koff];
  const float* b2p = &w2t[(32 + m) * W2T_STRIDE + koff];

  v8f acc0 = {0,0,0,0,0,0,0,0}, acc1 = {0,0,0,0,0,0,0,0}, acc2 = {0,0,0,0,0,0,0,0};
#pragma unroll
  for (int kk = 0; kk < 64; kk += 4) {
    v2f a  = *reinterpret_cast<const v2f*>(ap  + kk);
    v2f b0 = *reinterpret_cast<const v2f*>(b0p + kk);
    v2f b1 = *reinterpret_cast<const v2f*>(b1p + kk);
    v2f b2 = *reinterpret_cast<const v2f*>(b2p + kk);
    acc0 = __builtin_amdgcn_wmma_f32_16x16x4_f32(false, a, false, b0, (short)0, acc0, false, false);
    acc1 = __builtin_amdgcn_wmma_f32_16x16x4_f32(false, a, false, b1, (short)0, acc1, false, false);
    acc2 = __builtin_amdgcn_wmma_f32_16x16x4_f32(false, a, false, b2, (short)0, acc2, false, false);
  }

  int rbase = row0 + ((lane >> 4) << 3);
#pragma unroll
  for (int v = 0; v < 8; v++) {
    long long r = rbase + v;
    h2[r * NCLS + m]      = acc0[v];
    h2[r * NCLS + 16 + m] = acc1[v];
    if (32 + m < NCLS) h2[r * NCLS + 32 + m] = acc2[v];
  }
}

// ---- attention scalars ---------------------------------------------------
__global__ void k_att1(const float* __restrict__ h1, const float* __restrict__ as,
                       const float* __restrict__ ad, float* __restrict__ a1s,
                       float* __restrict__ a1d, int N) {
  int t = blockIdx.x * blockDim.x + threadIdx.x;
  if (t >= N * 8) return;
  int n = t >> 3, hd = t & 7;
  const float* hp = h1 + (long long)n * 64 + hd * 8;
  float s = 0.f, d = 0.f;
#pragma unroll
  for (int c = 0; c < 8; c++) {
    float v = hp[c];
    s += v * as[hd * 8 + c];
    d += v * ad[hd * 8 + c];
  }
  a1s[t] = s; a1d[t] = d;
}

__global__ void k_att2(const float* __restrict__ h2, const float* __restrict__ as,
                       const float* __restrict__ ad, float* __restrict__ a2s,
                       float* __restrict__ a2d, int N) {
  int n = blockIdx.x * blockDim.x + threadIdx.x;
  if (n >= N) return;
  const float* hp = h2 + (long long)n * NCLS;
  float s = 0.f, d = 0.f;
#pragma unroll
  for (int c = 0; c < NCLS; c++) {
    float v = hp[c];
    s += v * as[c];
    d += v * ad[c];
  }
  a2s[n] = s; a2d[n] = d;
}

// ---- layer-1 edge passes (8 heads) --------------------------------------
__global__ void k_max1(const long long* __restrict__ ei, long long E, int N,
                       const float* __restrict__ a1s, const float* __restrict__ a1d,
                       unsigned* __restrict__ m1) {
  long long t = (long long)blockIdx.x * blockDim.x + threadIdx.x;
  if (t >= (E + N) * 8) return;
  long long e = t >> 3; int hd = (int)(t & 7);
  int s, d; edge_nodes(ei, E, e, &s, &d);
  float sc = lrelu(a1s[s * 8 + hd] + a1d[d * 8 + hd]);
  atomicMax(&m1[d * 8 + hd], fkey(sc));
}

__global__ void k_sum1(const long long* __restrict__ ei, long long E, int N,
                       const float* __restrict__ a1s, const float* __restrict__ a1d,
                       const unsigned* __restrict__ m1, float* __restrict__ den1) {
  long long t = (long long)blockIdx.x * blockDim.x + threadIdx.x;
  if (t >= (E + N) * 8) return;
  long long e = t >> 3; int hd = (int)(t & 7);
  int s, d; edge_nodes(ei, E, e, &s, &d);
  float sc = lrelu(a1s[s * 8 + hd] + a1d[d * 8 + hd]);
  atomicAdd(&den1[d * 8 + hd], __expf(sc - funkey(m1[d * 8 + hd])));
}

__global__ void k_agg1(const long long* __restrict__ ei, long long E, int N,
                       const float* __restrict__ a1s, const float* __restrict__ a1d,
                       const unsigned* __restrict__ m1, const float* __restrict__ den1,
                       const float* __restrict__ h1, float* __restrict__ out1) {
  long long t = (long long)blockIdx.x * blockDim.x + threadIdx.x;
  if (t >= (E + N) * 8) return;
  long long e = t >> 3; int hd = (int)(t & 7);
  int s, d; edge_nodes(ei, E, e, &s, &d);
  float sc = lrelu(a1s[s * 8 + hd] + a1d[d * 8 + hd]);
  float alpha = __expf(sc - funkey(m1[d * 8 + hd])) / (den1[d * 8 + hd] + 1e-16f);
  const float* hp = h1 + (long long)s * 64 + hd * 8;
  float* op = out1 + (long long)d * 64 + hd * 8;
  v4f h0 = *reinterpret_cast<const v4f*>(hp);
  v4f h4 = *reinterpret_cast<const v4f*>(hp + 4);
  atomicAdd(op + 0, h0[0] * alpha); atomicAdd(op + 1, h0[1] * alpha);
  atomicAdd(op + 2, h0[2] * alpha); atomicAdd(op + 3, h0[3] * alpha);
  atomicAdd(op + 4, h4[0] * alpha); atomicAdd(op + 5, h4[1] * alpha);
  atomicAdd(op + 6, h4[2] * alpha); atomicAdd(op + 7, h4[3] * alpha);
}

// ---- bias + ELU (in place) ----------------------------------------------
__global__ void k_biaselu(float* __restrict__ o, const float* __restrict__ b1,
                          long long n) {
  long long t = (long long)blockIdx.x * blockDim.x + threadIdx.x;
  if (t >= n) return;
  float v = o[t] + b1[t & 63];
  o[t] = v > 0.f ? v : __expf(v) - 1.f;
}

// ---- layer-2 edge passes (1 head, 40 ch) --------------------------------
__global__ void k_max2(const long long* __restrict__ ei, long long E, int N,
                       const float* __restrict__ a2s, const float* __restrict__ a2d,
                       unsigned* __restrict__ m2) {
  long long e = (long long)blockIdx.x * blockDim.x + threadIdx.x;
  if (e >= E + N) return;
  int s, d; edge_nodes(ei, E, e, &s, &d);
  atomicMax(&m2[d], fkey(lrelu(a2s[s] + a2d[d])));
}

__global__ void k_sum2(const long long* __restrict__ ei, long long E, int N,
                       const float* __restrict__ a2s, const float* __restrict__ a2d,
                       const unsigned* __restrict__ m2, float* __restrict__ den2) {
  long long e = (long long)blockIdx.x * blockDim.x + threadIdx.x;
  if (e >= E + N) return;
  int s, d; edge_nodes(ei, E, e, &s, &d);
  float sc = lrelu(a2s[s] + a2d[d]);
  atomicAdd(&den2[d], __expf(sc - funkey(m2[d])));
}

__global__ void k_agg2(const long long* __restrict__ ei, long long E, int N,
                       const float* __restrict__ a2s, const float* __restrict__ a2d,
                       const unsigned* __restrict__ m2, const float* __restrict__ den2,
                       const float* __restrict__ h2, float* __restrict__ out) {
  long long e = (long long)blockIdx.x * blockDim.x + threadIdx.x;
  if (e >= E + N) return;
  int s, d; edge_nodes(ei, E, e, &s, &d);
  float sc = lrelu(a2s[s] + a2d[d]);
  float alpha = __expf(sc - funkey(m2[d])) / (den2[d] + 1e-16f);
  const float* hp = h2 + (long long)s * NCLS;
  float* op = out + (long long)d * NCLS;
#pragma unroll
  for (int c = 0; c < NCLS; c += 4) {
    v4f hv = *reinterpret_cast<const v4f*>(hp + c);
    atomicAdd(op + c + 0, hv[0] * alpha);
    atomicAdd(op + c + 1, hv[1] * alpha);
    atomicAdd(op + c + 2, hv[2] * alpha);
    atomicAdd(op + c + 3, hv[3] * alpha);
  }
}

// ---- final bias + softmax (in place on d_out) ---------------------------
__global__ void k_softmax(float* __restrict__ out, const float* __restrict__ b2,
                          int N) {
  int n = blockIdx.x * blockDim.x + threadIdx.x;
  if (n >= N) return;
  float* p = out + (long long)n * NCLS;
  float buf[NCLS];
  float mx = -3.4e38f;
#pragma unroll
  for (int c = 0; c < NCLS; c++) { float v = p[c] + b2[c]; buf[c] = v; mx = fmaxf(mx, v); }
  float sum = 0.f;
#pragma unroll
  for (int c = 0; c < NCLS; c++) { float e = __expf(buf[c] - mx); buf[c] = e; sum += e; }
  float inv = 1.f / sum;
#pragma unroll
  for (int c = 0; c < NCLS; c++) p[c] = buf[c] * inv;
}

// ---- launcher ------------------------------------------------------------
extern "C" void kernel_launch(void* const* d_in, const int* in_sizes, int n_in,
                              void* d_out, int out_size, void* d_ws, size_t ws_size,
                              hipStream_t stream) {
  const float*     x   = (const float*)d_in[0];
  const long long* ei  = (const long long*)d_in[1];
  const float*     W1  = (const float*)d_in[2];
  const float*     as1 = (const float*)d_in[3];
  const float*     ad1 = (const float*)d_in[4];
  const float*     b1  = (const float*)d_in[5];
  const float*     W2  = (const float*)d_in[6];
  const float*     as2 = (const float*)d_in[7];
  const float*     ad2 = (const float*)d_in[8];
  const float*     b2  = (const float*)d_in[9];
  float* out = (float*)d_out;

  const int       N = in_sizes[0] / 512;
  const long long E = (long long)in_sizes[1] / 2;

  // workspace layout (floats): h1[N*64] | out1[N*64] | a1s[N*8] | a1d[N*8]
  //                            | m1[N*8](u32) | den1[N*8] | a2s,a2d,m2,den2[N each]
  float*    ws   = (float*)d_ws;
  float*    h1   = ws;                          // reused as h2 later (N*40 <= N*64)
  float*    out1 = ws + (long long)N * 64;
  float*    a1s  = ws + (long long)N * 128;
  float*    a1d  = ws + (long long)N * 136;
  unsigned* m1   = (unsigned*)(ws + (long long)N * 144);
  float*    den1 = ws + (long long)N * 152;
  float*    a2s  = ws + (long long)N * 160;
  float*    a2d  = a2s + N;
  unsigned* m2   = (unsigned*)(a2s + 2LL * N);
  float*    den2 = a2s + 3LL * N;
  float*    h2   = h1;

  // zero accumulators / max keys (key 0 == below fkey(-inf)) and d_out
  k_zero<<<2048, 256, 0, stream>>>(out1, (long long)N * 100);       // out1..den2
  k_zero<<<512,  256, 0, stream>>>(a2s,  (long long)N * 4);         // layer-2 scalars
  k_zero<<<2048, 256, 0, stream>>>(out,  (long long)N * NCLS);

  int gblocks = (N / 16 + 7) / 8;
  k_gemm1<<<gblocks, 256, 0, stream>>>(x, W1, h1, N);
  k_att1<<<(N * 8 + 255) / 256, 256, 0, stream>>>(h1, as1, ad1, a1s, a1d, N);

  long long tot1 = (E + N) * 8;
  int eb8 = (int)((tot1 + 255) / 256);
  k_max1<<<eb8, 256, 0, stream>>>(ei, E, N, a1s, a1d, m1);
  k_sum1<<<eb8, 256, 0, stream>>>(ei, E, N, a1s, a1d, m1, den1);
  k_agg1<<<eb8, 256, 0, stream>>>(ei, E, N, a1s, a1d, m1, den1, h1, out1);

  k_biaselu<<<(int)(((long long)N * 64 + 255) / 256), 256, 0, stream>>>(out1, b1, (long long)N * 64);

  k_gemm2<<<gblocks, 256, 0, stream>>>(out1, W2, h2, N);
  k_att2<<<(N + 255) / 256, 256, 0, stream>>>(h2, as2, ad2, a2s, a2d, N);

  long long tot2 = E + N;
  int eb1 = (int)((tot2 + 255) / 256);
  k_max2<<<eb1, 256, 0, stream>>>(ei, E, N, a2s, a2d, m2);
  k_sum2<<<eb1, 256, 0, stream>>>(ei, E, N, a2s, a2d, m2, den2);
  k_agg2<<<eb1, 256, 0, stream>>>(ei, E, N, a2s, a2d, m2, den2, h2, out);

  k_softmax<<<(N + 255) / 256, 256, 0, stream>>>(out, b2, N);
}